// GCNRegressor_77275051590185
// MI455X (gfx1250) — compile-verified
//
#include <hip/hip_runtime.h>

#define F_IN 128
#define HDIM 64

typedef __attribute__((ext_vector_type(2))) float v2f;
typedef __attribute__((ext_vector_type(8))) float v8f;

// ---------------------------------------------------------------------------
// Degree accumulation: deg[dst] += 1 for each edge (self-loop added in k_dinv)
// ---------------------------------------------------------------------------
__global__ void k_deg(const long long* __restrict__ ei, float* __restrict__ deg, int E) {
    int e = blockIdx.x * blockDim.x + threadIdx.x;
    if (e < E) {
        long long d = ei[(size_t)E + e];
        atomicAdd(&deg[d], 1.0f);
    }
}

// dinv[n] = rsqrt(deg[n] + 1)   (deg includes self loop => always > 0)
__global__ void k_dinv(float* __restrict__ deg, int n) {
    int i = blockIdx.x * blockDim.x + threadIdx.x;
    if (i < n) {
        deg[i] = rsqrtf(deg[i] + 1.0f);
    }
}

// ---------------------------------------------------------------------------
// WMMA GEMM: out[N x 64] = X[N x K] @ W[K x 64]   (fp32, V_WMMA_F32_16X16X4_F32)
// Block = 256 threads = 8 waves, each wave computes a 16x64 output tile.
// W staged in LDS, K-pair interleaved: Wl2[(k/2)*64 + col] = {W[k][col], W[k+1][col]}
// so every B fragment is one contiguous ds_load_b64 into an aligned VGPR pair.
// ---------------------------------------------------------------------------
template <int K>
__global__ __launch_bounds__(256) void k_gemm(const float* __restrict__ X,
                                              const float* __restrict__ W,
                                              float* __restrict__ out, int nrows) {
    __shared__ float Wl[K * HDIM];
    for (int i = threadIdx.x; i < K * HDIM; i += 256) {
        int k = i / HDIM;
        int c = i - k * HDIM;
        Wl[(((k >> 1) * HDIM) + c) * 2 + (k & 1)] = W[i];
    }
    __syncthreads();
    const v2f* __restrict__ Wp = (const v2f*)Wl;   // Wp[kp*64 + col] = {W[2kp][col], W[2kp+1][col]}

    const int wave = threadIdx.x >> 5;
    const int lane = threadIdx.x & 31;
    const int row0 = (blockIdx.x * 8 + wave) * 16;
    if (row0 >= nrows) return;     // wave-uniform: EXEC stays all-ones for WMMA

    const int m  = lane & 15;      // M index (A) / N index (B,C)
    const int kh = lane >> 4;      // which K-half of the fragment this lane holds

    v8f acc0 = {0,0,0,0,0,0,0,0};
    v8f acc1 = {0,0,0,0,0,0,0,0};
    v8f acc2 = {0,0,0,0,0,0,0,0};
    v8f acc3 = {0,0,0,0,0,0,0,0};

    const float* __restrict__ xrow = X + (size_t)(row0 + m) * K;

    for (int k0 = 0; k0 < K; k0 += 4) {
        // A fragment: lane holds {X[row0+m][ka], X[row0+m][ka+1]}, ka = k0 + 2*kh
        v2f a = *(const v2f*)(xrow + k0 + 2 * kh);

        // B fragments: single b64 LDS load each, already paired for WMMA SRC1
        const v2f* __restrict__ wr = Wp + (size_t)((k0 >> 1) + kh) * HDIM;
        v2f b0 = wr[m];
        v2f b1 = wr[m + 16];
        v2f b2 = wr[m + 32];
        v2f b3 = wr[m + 48];

        acc0 = __builtin_amdgcn_wmma_f32_16x16x4_f32(false, a, false, b0, (short)0, acc0, false, false);
        acc1 = __builtin_amdgcn_wmma_f32_16x16x4_f32(false, a, false, b1, (short)0, acc1, false, false);
        acc2 = __builtin_amdgcn_wmma_f32_16x16x4_f32(false, a, false, b2, (short)0, acc2, false, false);
        acc3 = __builtin_amdgcn_wmma_f32_16x16x4_f32(false, a, false, b3, (short)0, acc3, false, false);
    }

    // C/D layout: VGPR r holds row r (lanes 0-15) / row 8+r (lanes 16-31), col = lane&15
    #pragma unroll
    for (int r = 0; r < 8; ++r) {
        float* orow = out + (size_t)(row0 + r + 8 * kh) * HDIM + m;
        orow[0]  = acc0[r];
        orow[16] = acc1[r];
        orow[32] = acc2[r];
        orow[48] = acc3[r];
    }
}

// ---------------------------------------------------------------------------
// Edge aggregation: agg[dst] += t[src] * dinv[src] * dinv[dst]
// One wave per edge, 2 features per lane -> 128B coalesced atomic bursts.
// Feature tables (25.6 MB) are L2-resident (192 MB L2); edge list is the only
// HBM-streamed data, so prefetch it ahead.
// ---------------------------------------------------------------------------
__global__ __launch_bounds__(256) void k_agg(const long long* __restrict__ ei,
                                             const float* __restrict__ dinv,
                                             const float* __restrict__ t,
                                             float* __restrict__ agg, int E) {
    const int wave = threadIdx.x >> 5;
    const int lane = threadIdx.x & 31;
    const int e = blockIdx.x * 8 + wave;
    if (e >= E) return;
    // stream-prefetch the edge list ~4KB ahead (global_prefetch_b8)
    __builtin_prefetch(ei + e + 512, 0, 1);
    __builtin_prefetch(ei + (size_t)E + e + 512, 0, 1);
    const long long s = ei[e];
    const long long d = ei[(size_t)E + e];
    const float nv = dinv[s] * dinv[d];
    const float* __restrict__ ts = t + (size_t)s * HDIM;
    float* __restrict__ ad = agg + (size_t)d * HDIM;
    atomicAdd(&ad[lane],      ts[lane]      * nv);
    atomicAdd(&ad[lane + 32], ts[lane + 32] * nv);
}

// ---------------------------------------------------------------------------
// Finalize: h = relu(agg + t * dinv^2 (self loop) + bias)
// ---------------------------------------------------------------------------
__global__ void k_finalize(const float* __restrict__ agg, const float* __restrict__ t,
                           const float* __restrict__ dinv, const float* __restrict__ b,
                           float* __restrict__ h, int n) {
    int idx = blockIdx.x * blockDim.x + threadIdx.x;
    int node = idx >> 6;
    int f = idx & 63;
    if (node < n) {
        float dv = dinv[node];
        float v = agg[idx] + t[idx] * dv * dv + b[f];
        h[idx] = v > 0.0f ? v : 0.0f;
    }
}

// ---------------------------------------------------------------------------
// Final linear: out[n] = dot(h[n,:], Wlin) + blin  (wave per node, shuffle reduce)
// ---------------------------------------------------------------------------
__global__ __launch_bounds__(256) void k_out(const float* __restrict__ h,
                                             const float* __restrict__ wl,
                                             const float* __restrict__ bl,
                                             float* __restrict__ out, int n) {
    const int wave = threadIdx.x >> 5;
    const int lane = threadIdx.x & 31;
    const int node = blockIdx.x * 8 + wave;
    if (node >= n) return;
    const float* __restrict__ hr = h + (size_t)node * HDIM;
    float p = hr[lane] * wl[lane] + hr[lane + 32] * wl[lane + 32];
    #pragma unroll
    for (int off = 16; off > 0; off >>= 1) p += __shfl_xor(p, off, 32);
    if (lane == 0) out[node] = p + bl[0];
}

// ---------------------------------------------------------------------------
extern "C" void kernel_launch(void* const* d_in, const int* in_sizes, int n_in,
                              void* d_out, int out_size, void* d_ws, size_t ws_size,
                              hipStream_t stream) {
    (void)n_in; (void)out_size; (void)ws_size;

    const float*     x    = (const float*)d_in[0];
    const long long* ei   = (const long long*)d_in[1];   // int64 edge_index [2, E]
    const float*     W1   = (const float*)d_in[2];
    const float*     b1   = (const float*)d_in[3];
    const float*     W2   = (const float*)d_in[4];
    const float*     b2   = (const float*)d_in[5];
    const float*     Wlin = (const float*)d_in[6];
    const float*     blin = (const float*)d_in[7];
    float* out = (float*)d_out;

    const int n = in_sizes[0] / F_IN;
    const int E = in_sizes[1] / 2;

    // workspace carve-up
    char* ws = (char*)d_ws;
    size_t o = 0;
    auto up = [](size_t v) { return (v + 255) & ~(size_t)255; };
    float* dinv = (float*)(ws + o); o += up((size_t)n * 4);
    float* t    = (float*)(ws + o); o += up((size_t)n * HDIM * 4);
    float* agg  = (float*)(ws + o); o += up((size_t)n * HDIM * 4);
    float* h    = (float*)(ws + o); o += up((size_t)n * HDIM * 4);

    const int NF = n * HDIM;

    // ---- normalization ----
    hipMemsetAsync(dinv, 0, (size_t)n * 4, stream);
    k_deg<<<(E + 255) / 256, 256, 0, stream>>>(ei, dinv, E);
    k_dinv<<<(n + 255) / 256, 256, 0, stream>>>(dinv, n);

    // ---- layer 1: t = x @ W1 ; agg = scatter ; h = relu(agg + self + b1) ----
    k_gemm<F_IN><<<(n + 127) / 128, 256, 0, stream>>>(x, W1, t, n);
    hipMemsetAsync(agg, 0, (size_t)NF * 4, stream);
    k_agg<<<(E + 7) / 8, 256, 0, stream>>>(ei, dinv, t, agg, E);
    k_finalize<<<(NF + 255) / 256, 256, 0, stream>>>(agg, t, dinv, b1, h, n);

    // ---- layer 2: t = h @ W2 ; agg = scatter ; h = relu(agg + self + b2) ----
    k_gemm<HDIM><<<(n + 127) / 128, 256, 0, stream>>>(h, W2, t, n);
    hipMemsetAsync(agg, 0, (size_t)NF * 4, stream);
    k_agg<<<(E + 7) / 8, 256, 0, stream>>>(ei, dinv, t, agg, E);
    k_finalize<<<(NF + 255) / 256, 256, 0, stream>>>(agg, t, dinv, b2, h, n);

    // ---- output head ----
    k_out<<<(n + 7) / 8, 256, 0, stream>>>(h, Wlin, blin, out, n);
}